// TemporalAttention_56762287784418
// MI455X (gfx1250) — compile-verified
//
#include <hip/hip_runtime.h>
#include <math.h>

// MI455X / gfx1250, wave32. fp32 WMMA 16x16x4 for the projection GEMMs.
typedef __attribute__((ext_vector_type(2))) float v2f;
typedef __attribute__((ext_vector_type(8))) float v8f;

#define TB   8
#define TT   16
#define THW  4096          // 64*64
#define TD   256
#define TDK  128
#define NPIX (TB * THW)    // 32768 pixels

// ---------------------------------------------------------------------------
// Kernel 1: precombine weights.
//   M   = W_q (256x128) @ W_k^T (128x256)   -> 256x256
//   Wvo = W_v (256x128) @ W_o   (128x256)   -> 256x256
// One wave per 16x16 output tile, K-loop of 16x16x4 fp32 WMMAs.
// A frag (16x4): lane l: m=l&15, kb=2*(l>>4); a={A[m,k+kb],A[m,k+kb+1]}
// B frag (4x16): lane l: n=l&15, kb=2*(l>>4); b={B[k+kb,n],B[k+kb+1,n]}
// C tile: vgpr r: lanes0-15 -> (M=r,N=l), lanes16-31 -> (M=8+r,N=l-16)
// ---------------------------------------------------------------------------
__global__ void __launch_bounds__(32)
build_mats(const float* __restrict__ Wq, const float* __restrict__ Wk,
           const float* __restrict__ Wv, const float* __restrict__ Wo,
           float* __restrict__ M, float* __restrict__ Wvo)
{
    const int lane = threadIdx.x & 31;
    const int row0 = (blockIdx.x >> 4) * 16;
    const int col0 = (blockIdx.x & 15) * 16;
    const int mn   = lane & 15;
    const int hi   = lane >> 4;
    const int kb   = hi * 2;

    v8f c = {0.f, 0.f, 0.f, 0.f, 0.f, 0.f, 0.f, 0.f};

    if (blockIdx.y == 0) {
        // M = Wq @ Wk^T : B[k][j] = Wk[j*128 + k] (contiguous in k -> v2f load)
        const float* ap = Wq + (size_t)(row0 + mn) * TDK + kb;
        const float* bp = Wk + (size_t)(col0 + mn) * TDK + kb;
        for (int k = 0; k < TDK; k += 4) {
            v2f a = *(const v2f*)(ap + k);
            v2f b = *(const v2f*)(bp + k);
            c = __builtin_amdgcn_wmma_f32_16x16x4_f32(false, a, false, b,
                                                      (short)0, c, false, false);
        }
#pragma unroll
        for (int r = 0; r < 8; ++r)
            M[(size_t)(row0 + r + 8 * hi) * TD + col0 + mn] = c[r];
    } else {
        // Wvo = Wv @ Wo : B[k][j] = Wo[k*256 + j]
        const float* ap = Wv + (size_t)(row0 + mn) * TDK + kb;
        for (int k = 0; k < TDK; k += 4) {
            v2f a = *(const v2f*)(ap + k);
            v2f b;
            b.x = Wo[(size_t)(k + kb)     * TD + col0 + mn];
            b.y = Wo[(size_t)(k + kb + 1) * TD + col0 + mn];
            c = __builtin_amdgcn_wmma_f32_16x16x4_f32(false, a, false, b,
                                                      (short)0, c, false, false);
        }
#pragma unroll
        for (int r = 0; r < 8; ++r)
            Wvo[(size_t)(row0 + r + 8 * hi) * TD + col0 + mn] = c[r];
    }
}

// ---------------------------------------------------------------------------
// Kernel 2/4: C[32768,256] = A[32768,256] @ Bm[256,256]  (fp32 WMMA).
// One wave owns a 16-row stripe and all 16 column tiles (128 accumulator
// VGPRs): A fragment loaded once per k-step, Bm stays cache-hot.
// Safe for in-place A==C: a wave reads only its own 16 rows, writes after.
// ---------------------------------------------------------------------------
__global__ void __launch_bounds__(256)
gemm_256(const float* __restrict__ A, const float* __restrict__ Bm,
         float* __restrict__ C)
{
    const int lane   = threadIdx.x & 31;
    const int wave   = threadIdx.x >> 5;
    const int stripe = blockIdx.x * 8 + wave;
    const int row0   = stripe * 16;
    const int mn     = lane & 15;
    const int hi     = lane >> 4;
    const int kb     = hi * 2;

    v8f acc[16];
#pragma unroll
    for (int i = 0; i < 16; ++i)
        acc[i] = (v8f){0.f, 0.f, 0.f, 0.f, 0.f, 0.f, 0.f, 0.f};

    const float* arow = A + (size_t)(row0 + mn) * TD + kb;
    for (int k = 0; k < TD; k += 4) {
        v2f a = *(const v2f*)(arow + k);
        const float* b0 = Bm + (size_t)(k + kb) * TD + mn;
#pragma unroll
        for (int ct = 0; ct < 16; ++ct) {
            v2f b;
            b.x = b0[ct * 16];
            b.y = b0[ct * 16 + TD];
            acc[ct] = __builtin_amdgcn_wmma_f32_16x16x4_f32(false, a, false, b,
                                                            (short)0, acc[ct],
                                                            false, false);
        }
    }
#pragma unroll
    for (int ct = 0; ct < 16; ++ct) {
#pragma unroll
        for (int r = 0; r < 8; ++r)
            C[(size_t)(row0 + r + 8 * hi) * TD + ct * 16 + mn] = acc[ct][r];
    }
}

// ---------------------------------------------------------------------------
// Kernel 3: single streaming pass over the history buffer.
// One wave per pixel: holds all 16 history rows (16 x 8 floats/lane = 128
// VGPRs), computes scores = (qm . x_t)/sqrt(DK), top-4 gate, decay-biased
// softmax, and xbar = sum_t attn_t * x_t.  xbar staged into d_out.
// ---------------------------------------------------------------------------
__global__ void __launch_bounds__(256)
attn_pixel(const float* __restrict__ hist, const float* __restrict__ qm,
           float* __restrict__ xbar)
{
    const int lane = threadIdx.x & 31;
    const int wave = threadIdx.x >> 5;
    const int p    = blockIdx.x * 8 + wave;
    const int b    = p >> 12;        // / THW
    const int hw   = p & (THW - 1);

    const float* q = qm + (size_t)p * TD + lane * 8;
    float4 q0 = *(const float4*)q;
    float4 q1 = *(const float4*)(q + 4);

    float4 X0[TT], X1[TT];
    float  sc[TT];
    const size_t base = ((size_t)b * TT * THW + hw) * TD + lane * 8;

#pragma unroll
    for (int t = 0; t < TT; ++t) {
        const float* xp = hist + base + (size_t)t * THW * TD;
        float4 x0 = *(const float4*)xp;
        float4 x1 = *(const float4*)(xp + 4);
        X0[t] = x0; X1[t] = x1;
        float pd = q0.x * x0.x + q0.y * x0.y + q0.z * x0.z + q0.w * x0.w
                 + q1.x * x1.x + q1.y * x1.y + q1.z * x1.z + q1.w * x1.w;
#pragma unroll
        for (int off = 16; off >= 1; off >>= 1)
            pd += __shfl_xor(pd, off, 32);
        sc[t] = pd * 0.08838834764831845f;   // 1/sqrt(128)
    }

    // top-4 gate over T (rank by strictly-greater count)
#pragma unroll
    for (int t = 0; t < TT; ++t) {
        int cnt = 0;
#pragma unroll
        for (int u = 0; u < TT; ++u) cnt += (sc[u] > sc[t]) ? 1 : 0;
        // decay bias log(0.95^(16-t) + 1e-10) == (16-t)*ln(0.95) to < 1 ulp
        float biased = sc[t] + (float)(TT - t) * -0.05129329438755058f;
        sc[t] = (cnt < 4) ? biased : -INFINITY;
    }

    float mx = sc[0];
#pragma unroll
    for (int t = 1; t < TT; ++t) mx = fmaxf(mx, sc[t]);
    float sum = 0.f;
#pragma unroll
    for (int t = 0; t < TT; ++t) { sc[t] = __expf(sc[t] - mx); sum += sc[t]; }
    const float inv = 1.f / sum;

    float4 a0 = {0.f, 0.f, 0.f, 0.f}, a1 = {0.f, 0.f, 0.f, 0.f};
#pragma unroll
    for (int t = 0; t < TT; ++t) {
        const float at = sc[t] * inv;
        a0.x += at * X0[t].x; a0.y += at * X0[t].y;
        a0.z += at * X0[t].z; a0.w += at * X0[t].w;
        a1.x += at * X1[t].x; a1.y += at * X1[t].y;
        a1.z += at * X1[t].z; a1.w += at * X1[t].w;
    }
    float* op = xbar + (size_t)p * TD + lane * 8;
    *(float4*)op       = a0;
    *(float4*)(op + 4) = a1;
}

// ---------------------------------------------------------------------------
extern "C" void kernel_launch(void* const* d_in, const int* in_sizes, int n_in,
                              void* d_out, int out_size, void* d_ws, size_t ws_size,
                              hipStream_t stream)
{
    const float* query = (const float*)d_in[0];   // [8,64,64,256]
    const float* hist  = (const float*)d_in[1];   // [8,16,64,64,256]
    const float* Wq    = (const float*)d_in[2];   // [256,128]
    const float* Wk    = (const float*)d_in[3];   // [256,128]
    const float* Wv    = (const float*)d_in[4];   // [256,128]
    const float* Wo    = (const float*)d_in[5];   // [128,256]
    float* out = (float*)d_out;                   // [8,64,64,256] == [32768,256]

    float* ws  = (float*)d_ws;
    float* M   = ws;                  // 256*256
    float* Wvo = ws + 65536;          // 256*256
    float* qm  = ws + 131072;         // 32768*256

    // 1) M = Wq Wk^T, Wvo = Wv Wo           (tiny, WMMA f32)
    build_mats<<<dim3(256, 2), 32, 0, stream>>>(Wq, Wk, Wv, Wo, M, Wvo);
    // 2) qm = query @ M                      (32 MB GEMM, WMMA f32)
    gemm_256<<<256, 256, 0, stream>>>(query, M, qm);
    // 3) single pass over 512 MiB history -> xbar (staged in d_out)
    attn_pixel<<<4096, 256, 0, stream>>>(hist, qm, out);
    // 4) out = xbar @ Wvo, in place          (WMMA f32)
    gemm_256<<<256, 256, 0, stream>>>(out, Wvo, out);
}